// MultiHeadGatLayer_26328149524692
// MI455X (gfx1250) — compile-verified
//
#include <hip/hip_runtime.h>
#include <stdint.h>

// MultiHeadGatLayer fused for MI455X (gfx1250, wave32).
// - WMMA f32 16x16x4 for all matmuls (full f32 precision path).
// - Flash-style attention: e/alpha (134M elems, 537MB) never hit memory.
// - ALL WMMA B-operands (z_B, WB_w, W_w) pre-packed in lane order ->
//   exactly one contiguous b64 load per lane per WMMA step.
// - Mask tile staged to LDS via Tensor Data Mover (D# with HW padding),
//   fenced with s_wait_tensorcnt; plain-load fallback otherwise.

typedef __attribute__((ext_vector_type(2))) float v2f;
typedef __attribute__((ext_vector_type(8))) float v8f;
typedef uint32_t u32x4 __attribute__((ext_vector_type(4)));
typedef int      i32x4 __attribute__((ext_vector_type(4)));
typedef int      i32x8 __attribute__((ext_vector_type(8)));

constexpr int NA = 2048, NB = 2048, INF_ = 128, H = 8, DH = 16;
constexpr float SLOPE = 0.01f;

__device__ __forceinline__ float leaky(float x) { return x >= 0.f ? x : SLOPE * x; }

// Padded-LDS mask layout (matches TDM pad: +1 DWORD per 256 DWORDs):
//   byte (r, c) lives at  r*2056 + c + ((c>>10)<<2)      (row bank = 2r)
constexpr int MROW = 2056;                 // bytes per padded mask row
constexpr int MWORDS = 16 * 514;           // dwords of LDS for the tile

// ---------------------------------------------------------------------------
// Kernel 0: fold attention vectors.  vX[h][i] = sum_d WX_w[h][i][d]*aX_w[h][d]
// ---------------------------------------------------------------------------
__global__ void prep_kernel(const float* __restrict__ WA_w, const float* __restrict__ WA_b,
                            const float* __restrict__ aA_w, const float* __restrict__ aA_b,
                            const float* __restrict__ WB_w, const float* __restrict__ WB_b,
                            const float* __restrict__ aB_w, const float* __restrict__ aB_b,
                            float* vA, float* cA, float* vB, float* cB) {
  const int t = threadIdx.x;
  for (int idx = t; idx < H * INF_; idx += blockDim.x) {
    const int h = idx >> 7, i = idx & 127;
    float sa = 0.f, sb = 0.f;
    for (int d = 0; d < DH; ++d) {
      sa += WA_w[(h * INF_ + i) * DH + d] * aA_w[h * DH + d];
      sb += WB_w[(h * INF_ + i) * DH + d] * aB_w[h * DH + d];
    }
    vA[idx] = sa;
    vB[idx] = sb;
  }
  if (t < H) {
    float sa = 0.f, sb = 0.f;
    for (int d = 0; d < DH; ++d) {
      sa += WA_b[t * DH + d] * aA_w[t * DH + d];
      sb += WB_b[t * DH + d] * aB_w[t * DH + d];
    }
    cA[t] = sa + aA_b[t];
    cB[t] = sb + aB_b[t];
  }
}

// ---------------------------------------------------------------------------
// Kernel 0b: pre-pack WMMA B-operands into lane order.
// Slot s of quad q (K = 4q..4q+3): s = lane*2+o, lane = half*16+d,
// holds element (K = 4q + 2*half + o, col = d).
//   WBp[w][q][s] = WB_w[w][4q+2*half+o][d]          (z_B projection weights)
//   Wp [w][q][s] = W_w[w*16+d][4q+2*half+o]         (output proj, transposed)
// ---------------------------------------------------------------------------
__global__ void pack_kernel(const float* __restrict__ WB_w, const float* __restrict__ W_w,
                            float* __restrict__ WBp, float* __restrict__ Wp) {
  const int idx = blockIdx.x * 256 + threadIdx.x;     // 0 .. 16383
  const int w = idx >> 11;                            // head / col-tile
  const int q = (idx >> 6) & 31;                      // quad (K/4)
  const int s = idx & 63;
  const int l = s >> 1, o = s & 1;
  const int half = l >> 4, d = l & 15;
  const int k = 4 * q + 2 * half + o;
  WBp[idx] = WB_w[(w * INF_ + k) * DH + d];
  Wp[idx]  = W_w[(size_t)(w * 16 + d) * 128 + k];
}

// ---------------------------------------------------------------------------
// Kernel 1: eX[b,h,n] = h_X[b,n,:] . vX[h] + cX[h]    (wave per row, wave32)
// ---------------------------------------------------------------------------
__global__ void escore_kernel(const float* __restrict__ hA, const float* __restrict__ hB,
                              const float* __restrict__ vA, const float* __restrict__ cA,
                              const float* __restrict__ vB, const float* __restrict__ cB,
                              float* eA, float* eB) {
  const int lane = threadIdx.x & 31;
  const int wave = threadIdx.x >> 5;
  const int row  = blockIdx.x * 8 + wave;          // b*2048 + n
  const int side = blockIdx.y;
  const float* src = side ? hB : hA;
  const float* v   = side ? vB : vA;
  const float* c   = side ? cB : cA;
  float*       e   = side ? eB : eA;
  const int b = row >> 11, n = row & 2047;
  const float4 x = *(const float4*)&src[(size_t)row * INF_ + lane * 4];
  for (int h = 0; h < H; ++h) {
    const float4 w = *(const float4*)&v[h * INF_ + lane * 4];
    float p = x.x * w.x + x.y * w.y + x.z * w.z + x.w * w.w;
    for (int off = 16; off; off >>= 1) p += __shfl_xor(p, off, 32);
    if (lane == 0) e[(b * H + h) * 2048 + n] = p + c[h];
  }
}

// ---------------------------------------------------------------------------
// Kernel 2: z_B projection, stored directly in WMMA-B lane order:
//   element (m, d)  ->  zBw[bh*32768 + (m>>2)*64 + (((m&3)>>1)*16 + d)*2 + (m&1)]
// B-operand weights come pre-packed (one b64 per lane per step).
// ---------------------------------------------------------------------------
__global__ void zb_kernel(const float* __restrict__ hB, const float* __restrict__ WBp,
                          const float* __restrict__ WB_b, float* __restrict__ zBw) {
  __shared__ float lds_h[16 * 132];
  const int t = threadIdx.x;
  const int g0 = blockIdx.x * 16;                  // first global row (b*2048+m)
  for (int idx = t; idx < 16 * INF_; idx += 256) {
    const int r = idx >> 7, k = idx & 127;
    lds_h[r * 132 + k] = hB[(size_t)(g0 + r) * INF_ + k];
  }
  __syncthreads();
  const int lane = t & 31, w = t >> 5;             // w = head
  const int half = lane >> 4, d = lane & 15;
  const v2f* wp = (const v2f*)(WBp + w * 2048) + lane;
  v8f c = {};
  for (int k0 = 0; k0 < INF_; k0 += 4) {
    const int k = k0 + 2 * half;
    v2f a;
    a.x = lds_h[d * 132 + k];
    a.y = lds_h[d * 132 + k + 1];
    const v2f bb = wp[(k0 >> 2) * 32];             // packed: one b64 / step
    c = __builtin_amdgcn_wmma_f32_16x16x4_f32(false, a, false, bb, (short)0, c, false, false);
  }
  const float bias = WB_b[w * DH + d];
  const int b = g0 >> 11, m0 = g0 & 2047;
  float* dst = zBw + (size_t)(b * H + w) * (NB * DH);
#pragma unroll
  for (int j = 0; j < 8; ++j) {
    const int m = m0 + j + half * 8;
    dst[(m >> 2) * 64 + ((m & 2) >> 1) * 32 + d * 2 + (m & 1)] = c[j] + bias;
  }
}

// ---------------------------------------------------------------------------
// Kernel 3: fused masked-leaky-softmax-attention.
// block = one (b, 16-row n-tile); 8 waves = 8 heads share the mask tile in LDS
// (loaded by the Tensor Data Mover with hardware padding when available).
// ---------------------------------------------------------------------------
__global__ void attn_kernel(const unsigned char* __restrict__ mask,
                            const float* __restrict__ eA, const float* __restrict__ eB,
                            const float* __restrict__ zBw, float* __restrict__ cat) {
  __shared__ uint32_t lds_mask[MWORDS];            // 16 rows, 2056B stride
  const int t = threadIdx.x;
  const int tile = blockIdx.x;                     // 0..511
  const int b = tile >> 7;
  const int n0 = (tile & 127) * 16;
  const unsigned char* gmask = mask + (size_t)(b * NA + n0) * NB;

#if __has_builtin(__builtin_amdgcn_tensor_load_to_lds)
  if (t < 32) {                                    // wave 0 issues the TDM op
    const uint64_t ga = (uint64_t)(uintptr_t)gmask;
    const uint32_t lo = (uint32_t)(uintptr_t)&lds_mask[0];   // LDS byte offset
    u32x4 g0v;                                     // D# group 0
    g0v[0] = 1u;                                   // count=1, user descriptor
    g0v[1] = lo;                                   // lds_addr
    g0v[2] = (uint32_t)ga;                         // global_addr[31:0]
    g0v[3] = (uint32_t)(ga >> 32) | 0x80000000u;   // addr[56:32] | type=2
    i32x8 g1v;                                     // D# group 1
    g1v[0] = (int)((1u << 20) | (7u << 22));       // pad_enable, 256DW interval, 1DW pad
    g1v[1] = (int)(2048u << 16);                   // tensor_dim0 = 2048 (bytes)
    g1v[2] = (int)(16u << 16);                     // tensor_dim1 = 16
    g1v[3] = (int)(2048u << 16);                   // tile_dim0 = 2048
    g1v[4] = 16;                                   // tile_dim1 = 16
    g1v[5] = 2048;                                 // tensor_dim0_stride = NB
    g1v[6] = 0;
    g1v[7] = 0;
    i32x4 z4 = {0, 0, 0, 0};
    i32x8 z8 = {0, 0, 0, 0, 0, 0, 0, 0};
    __builtin_amdgcn_tensor_load_to_lds(g0v, g1v, z4, z4, z8, 0);
    __builtin_amdgcn_s_wait_tensorcnt(0);
  }
#else
  {                                                // fallback: coalesced loads
    const uint32_t* msrc = (const uint32_t*)gmask;
    for (int idx = t; idx < 16 * 512; idx += 256) {
      const int r = idx >> 9, wo = idx & 511;
      lds_mask[r * 514 + wo + (wo >> 8)] = msrc[(size_t)r * 512 + wo];
    }
  }
#endif
  __syncthreads();

  const int lane = t & 31, w = t >> 5;             // w = head
  const int half = lane >> 4, rrow = lane & 15;
  const float myEA = eA[(b * H + w) * NA + n0 + rrow];
  const float* eBrow = &eB[(b * H + w) * NB];
  const unsigned char* mrow = (const unsigned char*)lds_mask + rrow * MROW;
  const v2f* zp = (const v2f*)(zBw + (size_t)(b * H + w) * (NB * DH)) + lane;

  float rs = 0.f;
  v8f c = {};
#pragma unroll 4
  for (int m0 = 0; m0 < NB; m0 += 4) {
    const int mi = m0 + half * 2;                  // this lane's K pair
    const float2 ev = *(const float2*)&eBrow[mi];
    const uint32_t moff = (uint32_t)mi + (((uint32_t)mi >> 10) << 2);
    const uint16_t mk = *(const uint16_t*)(mrow + moff);
    const float p0 = (mk & 0x00FF) ? __expf(leaky(myEA + ev.x)) : 0.f;
    const float p1 = (mk & 0xFF00) ? __expf(leaky(myEA + ev.y)) : 0.f;
    rs += p0 + p1;
    v2f a; a.x = p0; a.y = p1;
    const v2f bb = zp[(m0 >> 2) * 32];             // one coalesced b64 / step
    c = __builtin_amdgcn_wmma_f32_16x16x4_f32(false, a, false, bb, (short)0, c, false, false);
  }
  rs += __shfl_xor(rs, 16, 32);                    // lane holds rowsum(lane&15)
#pragma unroll
  for (int j = 0; j < 8; ++j) {
    const int r = j + half * 8;
    const float denom = __shfl(rs, r, 32);
    const float val = denom > 0.f ? c[j] / denom : 0.f;   // all-masked row -> 0
    cat[(size_t)(b * NA + n0 + r) * (H * DH) + w * DH + rrow] = val;
  }
}

// ---------------------------------------------------------------------------
// Kernel 4: out = cat @ W_w^T + W_b   (8192x128 @ 128x128), packed B-operand.
// ---------------------------------------------------------------------------
__global__ void out_kernel(const float* __restrict__ cat, const float* __restrict__ Wp,
                           const float* __restrict__ W_b, float* __restrict__ out) {
  __shared__ float lds_c[16 * 132];
  const int t = threadIdx.x;
  const int g0 = blockIdx.x * 16;
  for (int idx = t; idx < 16 * 128; idx += 256) {
    const int r = idx >> 7, k = idx & 127;
    lds_c[r * 132 + k] = cat[(size_t)(g0 + r) * 128 + k];
  }
  __syncthreads();
  const int lane = t & 31, w = t >> 5;             // w = column tile
  const int half = lane >> 4, d = lane & 15;
  const int o = w * 16 + d;
  const v2f* wp = (const v2f*)(Wp + w * 2048) + lane;
  v8f c = {};
  for (int k0 = 0; k0 < 128; k0 += 4) {
    const int k = k0 + 2 * half;
    v2f a;
    a.x = lds_c[d * 132 + k];
    a.y = lds_c[d * 132 + k + 1];
    const v2f bb = wp[(k0 >> 2) * 32];             // packed: one b64 / step
    c = __builtin_amdgcn_wmma_f32_16x16x4_f32(false, a, false, bb, (short)0, c, false, false);
  }
  const float bias = W_b[o];
#pragma unroll
  for (int j = 0; j < 8; ++j) {
    const int r = j + half * 8;
    out[(size_t)(g0 + r) * 128 + o] = c[j] + bias;
  }
}

// ---------------------------------------------------------------------------
extern "C" void kernel_launch(void* const* d_in, const int* in_sizes, int n_in,
                              void* d_out, int out_size, void* d_ws, size_t ws_size,
                              hipStream_t stream) {
  (void)in_sizes; (void)n_in; (void)out_size; (void)ws_size;
  const float* h_A = (const float*)d_in[0];
  const float* h_B = (const float*)d_in[1];
  const unsigned char* mask = (const unsigned char*)d_in[2];   // numpy bool bytes
  const float* WA_w = (const float*)d_in[3];
  const float* WA_b = (const float*)d_in[4];
  const float* WB_w = (const float*)d_in[5];
  const float* WB_b = (const float*)d_in[6];
  const float* aA_w = (const float*)d_in[7];
  const float* aA_b = (const float*)d_in[8];
  const float* aB_w = (const float*)d_in[9];
  const float* aB_b = (const float*)d_in[10];
  const float* W_w  = (const float*)d_in[11];
  const float* W_b  = (const float*)d_in[12];
  float* out = (float*)d_out;

  float* ws  = (float*)d_ws;
  float* vA  = ws;                  // 1024
  float* cA  = ws + 1024;           // 8
  float* vB  = ws + 1040;           // 1024
  float* cB  = ws + 2064;           // 8
  float* WBp = ws + 2080;           // 16384 (packed z_B-projection weights)
  float* Wp  = WBp + 16384;         // 16384 (packed output-projection weights)
  float* eA  = Wp + 16384;          // 65536
  float* eB  = eA + 65536;          // 65536
  float* zBw = eB + 65536;          // 1048576 (WMMA-B lane-ordered z_B)
  float* cat = zBw + 1048576;       // 1048576   (total ~9 MB of d_ws)

  prep_kernel<<<1, 256, 0, stream>>>(WA_w, WA_b, aA_w, aA_b, WB_w, WB_b, aB_w, aB_b,
                                     vA, cA, vB, cB);
  pack_kernel<<<64, 256, 0, stream>>>(WB_w, W_w, WBp, Wp);
  escore_kernel<<<dim3(1024, 2), 256, 0, stream>>>(h_A, h_B, vA, cA, vB, cB, eA, eB);
  zb_kernel<<<512, 256, 0, stream>>>(h_B, WBp, WB_b, zBw);
  attn_kernel<<<512, 256, 0, stream>>>(mask, eA, eB, zBw, cat);
  out_kernel<<<512, 256, 0, stream>>>(cat, Wp, W_b, out);
}